// LomaSumAggr_69990787056179
// MI455X (gfx1250) — compile-verified
//
#include <hip/hip_runtime.h>

// Segment-sum (scatter-add) of a [1.6M x 64] f32 matrix into 50K segments.
//
// Memory-bound: ~430 MB of HBM traffic -> ~18us floor at 23.3 TB/s.
// The 12.8 MB output lives in L2 (192 MB); all 102.4M f32 atomic adds resolve
// there. Mapping: one wave32 per row; lane l covers columns l and l+32, so
// each global_atomic_add_f32 instruction touches 32 consecutive dwords (two
// full 64B lines of one segment row) -> maximal L2 atomic coalescing.
// Input is streamed with TH_LOAD_NT so it does not evict the accumulation
// working set from L2; global_prefetch_b8 provides a 512B lookahead.

#define N_COLS 64
#define ROWS_PER_WAVE 8
#define WAVES_PER_BLOCK 8
#define BLOCK_THREADS (WAVES_PER_BLOCK * 32)

__global__ __launch_bounds__(BLOCK_THREADS) void scatter_add_rows(
    const float* __restrict__ in, const int* __restrict__ idx,
    float* __restrict__ out, int n_rows) {
  const int lane = threadIdx.x & 31;  // wave32
  const long long gwave =
      (long long)blockIdx.x * WAVES_PER_BLOCK + (threadIdx.x >> 5);
  long long row = gwave * (long long)ROWS_PER_WAVE;
  if (row >= n_rows) return;
  long long row_end = row + ROWS_PER_WAVE;
  if (row_end > n_rows) row_end = n_rows;

  for (; row < row_end; ++row) {
    // Speculative L2 prefetch two rows (512 B) ahead, clamped to the
    // allocation. Emits global_prefetch_b8 on gfx1250.
    const long long pr = (row + 2 < n_rows) ? (row + 2) : row;
    __builtin_prefetch(in + pr * N_COLS + lane, 0, 1);

    // Segment id is wave-uniform (one wave per row): force it into an SGPR.
    const int seg = __builtin_amdgcn_readfirstlane(idx[row]);

    // Stream the row non-temporally: 2 x 128B coalesced loads per wave.
    const float* rp = in + row * (long long)N_COLS;
    const float a = __builtin_nontemporal_load(rp + lane);
    const float b = __builtin_nontemporal_load(rp + lane + 32);

    // Two returnless global_atomic_add_f32 instructions per row; each covers
    // 32 consecutive dwords of out[seg] (two full 64B L2 lines).
    float* op = out + (long long)seg * N_COLS;
    unsafeAtomicAdd(op + lane, a);
    unsafeAtomicAdd(op + lane + 32, b);
  }
}

__global__ __launch_bounds__(256) void zero_f4(float4* __restrict__ out, int n4) {
  int i = blockIdx.x * blockDim.x + threadIdx.x;
  if (i < n4) out[i] = make_float4(0.f, 0.f, 0.f, 0.f);
}

extern "C" void kernel_launch(void* const* d_in, const int* in_sizes, int n_in,
                              void* d_out, int out_size, void* d_ws, size_t ws_size,
                              hipStream_t stream) {
  const float* in = (const float*)d_in[0];
  const int* idx = (const int*)d_in[1];
  float* out = (float*)d_out;

  const int n_rows = in_sizes[0] / N_COLS;

  // Zero the (poisoned) output before accumulation; stream ordering makes
  // this safe w.r.t. the scatter kernel.
  const int n4 = out_size / 4;
  const int zb = (n4 + 255) / 256;
  zero_f4<<<zb, 256, 0, stream>>>((float4*)out, n4);

  const long long waves =
      ((long long)n_rows + ROWS_PER_WAVE - 1) / ROWS_PER_WAVE;
  const int blocks = (int)((waves + WAVES_PER_BLOCK - 1) / WAVES_PER_BLOCK);
  scatter_add_rows<<<blocks, BLOCK_THREADS, 0, stream>>>(in, idx, out, n_rows);
}